// ModelHeterogene_65549790872160
// MI455X (gfx1250) — compile-verified
//
#include <hip/hip_runtime.h>

typedef __attribute__((ext_vector_type(2))) float v2f;
typedef __attribute__((ext_vector_type(8))) float v8f;

#define NN 100000
#define EE 1600000
#define HH 128
#define CHH 16
#define PP 10
#define RBLK 512
#define SBLK 512

// ---------------- utility ----------------
__global__ void k_fill(float* __restrict__ p, long long n, float v) {
  long long i = (long long)blockIdx.x * blockDim.x + threadIdx.x;
  if (i < n) p[i] = v;
}

// deg[i] starts at 1.0 (self loop); add edge weights by row
__global__ void k_deg(const int* __restrict__ row, const float* __restrict__ w,
                      float* __restrict__ deg, int e) {
  int i = blockIdx.x * blockDim.x + threadIdx.x;
  if (i < e) atomicAdd(&deg[row[i]], w[i]);
}

__global__ void k_dis(float* __restrict__ d, int n) {
  int i = blockIdx.x * blockDim.x + threadIdx.x;
  if (i < n) { float v = d[i]; d[i] = (v > 0.f) ? rsqrtf(v) : 0.f; }
}

__global__ void k_norm(const int* __restrict__ row, const int* __restrict__ col,
                       const float* __restrict__ w, const float* __restrict__ dis,
                       float* __restrict__ nrm, int e) {
  int i = blockIdx.x * blockDim.x + threadIdx.x;
  if (i < e) nrm[i] = dis[row[i]] * w[i] * dis[col[i]];
}

// ---------------- WMMA GEMM: C[N x 128] = A[N x 128] @ W[128 x 128] ----------------
// One wave computes a 16x128 output stripe; A fragment reused across 8 N-tiles.
__global__ void k_gemm128(const float* __restrict__ A, const float* __restrict__ W,
                          float* __restrict__ C, int ntiles) {
  int tile = blockIdx.x * (blockDim.x >> 5) + (threadIdx.x >> 5);
  if (tile >= ntiles) return;
  int lane = threadIdx.x & 31;
  int half = lane >> 4;
  int l16  = lane & 15;
  int row0 = tile * 16;
  v8f acc[8] = {};
  const float* arow = A + (size_t)(row0 + l16) * HH + 2 * half;
  for (int k = 0; k < HH; k += 4) {
    v2f a;
    a.x = arow[k];
    a.y = arow[k + 1];
    const float* wp = W + (size_t)(k + 2 * half) * HH + l16;
#pragma unroll
    for (int t = 0; t < 8; ++t) {
      v2f b;
      b.x = wp[t * 16];
      b.y = wp[t * 16 + HH];
      acc[t] = __builtin_amdgcn_wmma_f32_16x16x4_f32(
          false, a, false, b, (short)0, acc[t], false, false);
    }
  }
#pragma unroll
  for (int t = 0; t < 8; ++t) {
    int n = t * 16 + l16;
#pragma unroll
    for (int r = 0; r < 8; ++r)
      C[(size_t)(row0 + half * 8 + r) * HH + n] = acc[t][r];
  }
}

// ---------------- edge scatter-gather: out[row] += norm * g[col] ----------------
__global__ void k_aggregate(const float* __restrict__ g, const int* __restrict__ row,
                            const int* __restrict__ col, const float* __restrict__ nrm,
                            float* __restrict__ out, int e_cnt) {
  long long tid = (long long)blockIdx.x * blockDim.x + threadIdx.x;
  if (tid >= (long long)e_cnt * 32) return;
  int e = (int)(tid >> 5);
  int q = (int)(tid & 31);
  int r = row[e], c = col[e];
  float w = nrm[e];
  const float4 v = ((const float4*)(g + (size_t)c * HH))[q];
  float* op = out + (size_t)r * HH + q * 4;
  atomicAdd(op + 0, w * v.x);
  atomicAdd(op + 1, w * v.y);
  atomicAdd(op + 2, w * v.z);
  atomicAdd(op + 3, w * v.w);
}

// h = relu(h + dis^2 * g + b)   (self-loop term + bias + relu, in place)
__global__ void k_bias_relu_self(float* __restrict__ h, const float* __restrict__ g,
                                 const float* __restrict__ dis, const float* __restrict__ b,
                                 int n) {
  long long tid = (long long)blockIdx.x * blockDim.x + threadIdx.x;
  if (tid >= (long long)n * HH) return;
  int i = (int)(tid >> 7);
  int t = (int)(tid & 127);
  float dd = dis[i];
  float v = h[tid] + dd * dd * g[tid] + b[t];
  h[tid] = fmaxf(v, 0.f);
}

// ---------------- fused logits: relu(h@W1+b1)@w2 + b2 ----------------
__global__ void k_mlp_logits(const float* __restrict__ A, const float* __restrict__ W1,
                             const float* __restrict__ B1, const float* __restrict__ W2,
                             const float* __restrict__ B2, float* __restrict__ logits,
                             int ntiles) {
  int tile = blockIdx.x * (blockDim.x >> 5) + (threadIdx.x >> 5);
  if (tile >= ntiles) return;
  int lane = threadIdx.x & 31;
  int half = lane >> 4;
  int l16  = lane & 15;
  int row0 = tile * 16;
  v8f acc[8] = {};
  const float* arow = A + (size_t)(row0 + l16) * HH + 2 * half;
  for (int k = 0; k < HH; k += 4) {
    v2f a;
    a.x = arow[k];
    a.y = arow[k + 1];
    const float* wp = W1 + (size_t)(k + 2 * half) * HH + l16;
#pragma unroll
    for (int t = 0; t < 8; ++t) {
      v2f b;
      b.x = wp[t * 16];
      b.y = wp[t * 16 + HH];
      acc[t] = __builtin_amdgcn_wmma_f32_16x16x4_f32(
          false, a, false, b, (short)0, acc[t], false, false);
    }
  }
  // epilogue: relu(acc + b1) dotted with w2, reduced across the 16 lanes of each half
  float part[8];
#pragma unroll
  for (int r = 0; r < 8; ++r) part[r] = 0.f;
#pragma unroll
  for (int t = 0; t < 8; ++t) {
    int n = t * 16 + l16;
    float w2 = W2[n];
    float b1 = B1[n];
#pragma unroll
    for (int r = 0; r < 8; ++r)
      part[r] += fmaxf(acc[t][r] + b1, 0.f) * w2;
  }
#pragma unroll
  for (int m = 1; m < 16; m <<= 1)
#pragma unroll
    for (int r = 0; r < 8; ++r)
      part[r] += __shfl_xor(part[r], m, 32);
  if (l16 == 0) {
    float b2 = B2[0];
#pragma unroll
    for (int r = 0; r < 8; ++r)
      logits[row0 + half * 8 + r] = part[r] + b2;
  }
}

// ---------------- column-wise mean + masked max over rows ----------------
__global__ void k_reduce_mm(const float* __restrict__ h, const int* __restrict__ ready,
                            float* __restrict__ psum, float* __restrict__ pmax) {
  int t = threadIdx.x;  // 0..127 (feature)
  float s = 0.f, m = -__builtin_inff();
  for (int i = blockIdx.x; i < NN; i += gridDim.x) {
    float v = h[(size_t)i * HH + t];
    s += v;
    if (ready[i] > 0) m = fmaxf(m, v);
  }
  psum[blockIdx.x * HH + t] = s;
  pmax[blockIdx.x * HH + t] = m;
}

__global__ void k_final_mm(const float* __restrict__ psum, const float* __restrict__ pmax,
                           float* __restrict__ small, int nblk) {
  int t = threadIdx.x;  // 0..127
  float s = 0.f, m = -__builtin_inff();
  for (int b = 0; b < nblk; ++b) {
    s += psum[b * HH + t];
    m = fmaxf(m, pmax[b * HH + t]);
  }
  small[t] = s / (float)NN;   // mean
  small[HH + t] = m;          // masked max
}

// ---------------- tiny head: cluster linear, pass MLP, value ----------------
// small layout: [0..127]=mean [128..255]=xmax [400]=pass_logit [401]=gmax [402]=gsum
__global__ void k_head(float* __restrict__ small, const float* __restrict__ pe,
                       const float* __restrict__ cl_w, const float* __restrict__ cl_b,
                       const float* __restrict__ pw1, const float* __restrict__ pb1,
                       const float* __restrict__ pw2, const float* __restrict__ pb2,
                       const float* __restrict__ vw, const float* __restrict__ vb,
                       float* __restrict__ vout) {
  __shared__ float cat[HH + CHH];
  __shared__ float hid[HH];
  __shared__ float red[256];
  int t = threadIdx.x;
  if (t < HH) cat[t] = small[HH + t];  // masked max part
  if (t < CHH) {
    float s = cl_b[t];
    for (int i = 0; i < 3 * PP; ++i) s += pe[i] * cl_w[i * CHH + t];
    cat[HH + t] = s;
  }
  __syncthreads();
  if (t < HH) {
    float s = pb1[t];
    for (int i = 0; i < HH + CHH; ++i) s += cat[i] * pw1[i * HH + t];
    hid[t] = fmaxf(s, 0.f);
  }
  __syncthreads();
  red[t] = (t < HH) ? hid[t] * pw2[t] : 0.f;
  __syncthreads();
  for (int s = 128; s > 0; s >>= 1) { if (t < s) red[t] += red[t + s]; __syncthreads(); }
  if (t == 0) small[400] = red[0] + pb2[0];
  __syncthreads();
  red[t] = (t < HH) ? small[t] * vw[t] : 0.f;
  __syncthreads();
  for (int s = 128; s > 0; s >>= 1) { if (t < s) red[t] += red[t + s]; __syncthreads(); }
  if (t == 0) vout[0] = red[0] + vb[0];
}

// ---------------- masked softmax over [N ready logits, pass_logit] ----------------
__global__ void k_smax(const float* __restrict__ logits, const int* __restrict__ ready,
                       float* __restrict__ pred) {
  __shared__ float sm[256];
  int t = threadIdx.x;
  float m = -__builtin_inff();
  for (int i = blockIdx.x * 256 + t; i < NN; i += gridDim.x * 256)
    if (ready[i] > 0) m = fmaxf(m, logits[i]);
  sm[t] = m; __syncthreads();
  for (int s = 128; s > 0; s >>= 1) { if (t < s) sm[t] = fmaxf(sm[t], sm[t + s]); __syncthreads(); }
  if (t == 0) pred[blockIdx.x] = sm[0];
}

__global__ void k_smax_fin(const float* __restrict__ pred, int n, float* __restrict__ small) {
  __shared__ float sm[256];
  int t = threadIdx.x;
  float m = -__builtin_inff();
  for (int i = t; i < n; i += 256) m = fmaxf(m, pred[i]);
  sm[t] = m; __syncthreads();
  for (int s = 128; s > 0; s >>= 1) { if (t < s) sm[t] = fmaxf(sm[t], sm[t + s]); __syncthreads(); }
  if (t == 0) small[401] = fmaxf(sm[0], small[400]);
}

__global__ void k_ssum(const float* __restrict__ logits, const int* __restrict__ ready,
                       const float* __restrict__ small, float* __restrict__ pred) {
  __shared__ float sm[256];
  int t = threadIdx.x;
  float gm = small[401];
  float s = 0.f;
  for (int i = blockIdx.x * 256 + t; i < NN; i += gridDim.x * 256)
    if (ready[i] > 0) s += expf(logits[i] - gm);
  sm[t] = s; __syncthreads();
  for (int q = 128; q > 0; q >>= 1) { if (t < q) sm[t] += sm[t + q]; __syncthreads(); }
  if (t == 0) pred[blockIdx.x] = sm[0];
}

__global__ void k_ssum_fin(const float* __restrict__ pred, int n, float* __restrict__ small) {
  __shared__ float sm[256];
  int t = threadIdx.x;
  float s = 0.f;
  for (int i = t; i < n; i += 256) s += pred[i];
  sm[t] = s; __syncthreads();
  for (int q = 128; q > 0; q >>= 1) { if (t < q) sm[t] += sm[t + q]; __syncthreads(); }
  if (t == 0) small[402] = sm[0] + expf(small[400] - small[401]);
}

__global__ void k_probs(const float* __restrict__ logits, const int* __restrict__ ready,
                        const float* __restrict__ small, float* __restrict__ out) {
  int i = blockIdx.x * blockDim.x + threadIdx.x;
  float gm = small[401];
  float inv = 1.f / small[402];
  if (i < NN)
    out[i] = (ready[i] > 0) ? expf(logits[i] - gm) * inv : 0.f;
  if (i == 0)
    out[NN] = expf(small[400] - gm) * inv;
}

// ---------------- launch ----------------
extern "C" void kernel_launch(void* const* d_in, const int* in_sizes, int n_in,
                              void* d_out, int out_size, void* d_ws, size_t ws_size,
                              hipStream_t stream) {
  const float* x      = (const float*)d_in[0];
  const int*   edges  = (const int*)d_in[1];
  const float* ea     = (const float*)d_in[2];
  const int*   ready  = (const int*)d_in[3];
  const float* pe     = (const float*)d_in[4];
  const float* gcn_w  = (const float*)d_in[5];
  const float* gcn_b  = (const float*)d_in[6];
  const float* mlp_w1 = (const float*)d_in[7];
  const float* mlp_b1 = (const float*)d_in[8];
  const float* mlp_w2 = (const float*)d_in[9];
  const float* mlp_b2 = (const float*)d_in[10];
  const float* val_w  = (const float*)d_in[11];
  const float* val_b  = (const float*)d_in[12];
  const float* cl_w   = (const float*)d_in[13];
  const float* cl_b   = (const float*)d_in[14];
  const float* pw1    = (const float*)d_in[15];
  const float* pb1    = (const float*)d_in[16];
  const float* pw2    = (const float*)d_in[17];
  const float* pb2    = (const float*)d_in[18];
  float* dout = (float*)d_out;

  const int* rowp = edges;
  const int* colp = edges + EE;

  float* wsf = (float*)d_ws;
  size_t o = 0;
  float* bufA   = wsf + o; o += (size_t)NN * HH;
  float* bufB   = wsf + o; o += (size_t)NN * HH;
  float* disv   = wsf + o; o += NN;
  float* nrmv   = wsf + o; o += EE;
  float* logits = wsf + o; o += NN;
  float* psum   = wsf + o; o += (size_t)RBLK * HH;
  float* pmax   = wsf + o; o += (size_t)RBLK * HH;
  float* sred   = wsf + o; o += SBLK;
  float* sred2  = wsf + o; o += SBLK;
  float* small  = wsf + o; o += 512;

  const long long nh = (long long)NN * HH;
  const int gNH  = (int)((nh + 255) / 256);
  const int gN   = (NN + 255) / 256;
  const int gE   = (EE + 255) / 256;
  const int gAgg = (int)(((long long)EE * 32 + 255) / 256);
  const int ntiles = NN / 16;            // 6250
  const int gGemm  = (ntiles + 7) / 8;   // 8 waves per block

  // normalization coefficients (shared by both GCN layers)
  k_fill<<<gN, 256, 0, stream>>>(disv, NN, 1.0f);
  k_deg<<<gE, 256, 0, stream>>>(rowp, ea, disv, EE);
  k_dis<<<gN, 256, 0, stream>>>(disv, NN);
  k_norm<<<gE, 256, 0, stream>>>(rowp, colp, ea, disv, nrmv, EE);

  // GCN layer 1
  k_gemm128<<<gGemm, 256, 0, stream>>>(x, gcn_w, bufA, ntiles);
  k_fill<<<gNH, 256, 0, stream>>>(bufB, nh, 0.f);
  k_aggregate<<<gAgg, 256, 0, stream>>>(bufA, rowp, colp, nrmv, bufB, EE);
  k_bias_relu_self<<<gNH, 256, 0, stream>>>(bufB, bufA, disv, gcn_b, NN);

  // GCN layer 2
  k_gemm128<<<gGemm, 256, 0, stream>>>(bufB, gcn_w + HH * HH, bufA, ntiles);
  k_fill<<<gNH, 256, 0, stream>>>(bufB, nh, 0.f);
  k_aggregate<<<gAgg, 256, 0, stream>>>(bufA, rowp, colp, nrmv, bufB, EE);
  k_bias_relu_self<<<gNH, 256, 0, stream>>>(bufB, bufA, disv, gcn_b + HH, NN);

  // node logits (fused MLP)
  k_mlp_logits<<<gGemm, 256, 0, stream>>>(bufB, mlp_w1, mlp_b1, mlp_w2, mlp_b2,
                                          logits, ntiles);

  // mean / masked-max reductions
  k_reduce_mm<<<RBLK, 128, 0, stream>>>(bufB, ready, psum, pmax);
  k_final_mm<<<1, 128, 0, stream>>>(psum, pmax, small, RBLK);

  // pass branch + value head (v -> dout[N+1])
  k_head<<<1, 256, 0, stream>>>(small, pe, cl_w, cl_b, pw1, pb1, pw2, pb2,
                                val_w, val_b, dout + NN + 1);

  // masked softmax over N+1 entries -> dout[0..N]
  k_smax<<<SBLK, 256, 0, stream>>>(logits, ready, sred);
  k_smax_fin<<<1, 256, 0, stream>>>(sred, SBLK, small);
  k_ssum<<<SBLK, 256, 0, stream>>>(logits, ready, small, sred2);
  k_ssum_fin<<<1, 256, 0, stream>>>(sred2, SBLK, small);
  k_probs<<<gN, 256, 0, stream>>>(logits, ready, small, dout);
}